// CorefHead_76957224010078
// MI455X (gfx1250) — compile-verified
//
#include <hip/hip_runtime.h>
#include <hip/hip_bf16.h>

#define B_    64
#define S_    512
#define H_    1024
#define HID_  512
#define K3H_  3072
#define EPS_  1e-5f

typedef __attribute__((ext_vector_type(2))) float v2f;
typedef __attribute__((ext_vector_type(4))) float v4f;
typedef __attribute__((ext_vector_type(8))) float v8f;

// ---------------------------------------------------------------------------
// Phase 1: span means + pron gather -> x (B x 3H) in workspace.
// HBM-bound phase (~90MB of reads). 256 threads/block, each owning 4
// consecutive columns -> one float4 (global_load_b128) per row per thread,
// a single pass over the span rows, 2-row unroll for MLP.
// ---------------------------------------------------------------------------
__global__ void coref_span_gather(const float* __restrict__ bert,
                                  const int* __restrict__ offsets,
                                  float* __restrict__ x) {
    const int b = blockIdx.x;
    const int* off = offsets + b * 5;
    const int a0 = off[0], a1 = off[1];
    const int c0 = off[2], c1 = off[3];
    const int pr = off[4];
    const float inv_a = 1.0f / (float)(a1 - a0 + 1);
    const float inv_c = 1.0f / (float)(c1 - c0 + 1);

    // Row stride in float4 units: H/4 = 256. threadIdx.x in [0,256).
    const v4f* base = (const v4f*)(bert + (size_t)b * S_ * H_);
    const int t = threadIdx.x;

    // --- span A mean ---
    v4f sa0 = {0.f, 0.f, 0.f, 0.f}, sa1 = {0.f, 0.f, 0.f, 0.f};
    {
        int s = a0;
        for (; s + 1 <= a1; s += 2) {
            v4f r0 = base[(size_t)s * 256 + t];
            v4f r1 = base[(size_t)(s + 1) * 256 + t];
            sa0 += r0;
            sa1 += r1;
        }
        if (s <= a1) sa0 += base[(size_t)s * 256 + t];
    }
    v4f ea = (sa0 + sa1) * inv_a;

    // --- span B mean ---
    v4f sc0 = {0.f, 0.f, 0.f, 0.f}, sc1 = {0.f, 0.f, 0.f, 0.f};
    {
        int s = c0;
        for (; s + 1 <= c1; s += 2) {
            v4f r0 = base[(size_t)s * 256 + t];
            v4f r1 = base[(size_t)(s + 1) * 256 + t];
            sc0 += r0;
            sc1 += r1;
        }
        if (s <= c1) sc0 += base[(size_t)s * 256 + t];
    }
    v4f ec = (sc0 + sc1) * inv_c;

    // --- pron row gather ---
    v4f ep = base[(size_t)pr * 256 + t];

    v4f* xb = (v4f*)(x + (size_t)b * K3H_);
    xb[t]       = ea;   // emb_a -> x[b][0:1024]
    xb[256 + t] = ec;   // emb_b -> x[b][1024:2048]
    xb[512 + t] = ep;   // emb_p -> x[b][2048:3072]
}

// ---------------------------------------------------------------------------
// Phase 2: h = LeakyReLU(BN(x @ W1 + b1)) via V_WMMA_F32_16X16X4_F32.
// One wave32 per 16x16 output tile. M=64 -> 4 tiles, N=512 -> 32 tiles,
// K=3072 stepped by 4.
//
// fp32 A 16x4 layout (ISA 7.12.2): lanes 0-15 hold M=lane, VGPR0=K0 VGPR1=K1;
// lanes 16-31 hold M=lane-16, VGPR0=K2 VGPR1=K3. B (4x16) mirrors with N in
// place of M. C/D 16x16 f32: VGPR i, lanes 0-15 -> M=i, lanes 16-31 -> M=i+8,
// N = lane%16.
// ---------------------------------------------------------------------------
__global__ void coref_gemm1_wmma(const float* __restrict__ x,
                                 const float* __restrict__ W1,
                                 const float* __restrict__ b1,
                                 const float* __restrict__ gamma,
                                 const float* __restrict__ beta,
                                 const float* __restrict__ rmean,
                                 const float* __restrict__ rvar,
                                 float* __restrict__ h) {
    const int wave = threadIdx.x >> 5;
    const int lane = threadIdx.x & 31;
    const int tile = blockIdx.x * (blockDim.x >> 5) + wave;   // 0..127
    const int mt = tile & 3;      // 4 tiles over M=64
    const int nt = tile >> 2;     // 32 tiles over N=512
    const int m0 = mt * 16;
    const int n0 = nt * 16;
    const int half = lane >> 4;   // 0: K pair {0,1}; 1: K pair {2,3}
    const int l16  = lane & 15;

    // A: row (m0+l16) of x, starting at column offset 2*half
    const float* arow = x + (size_t)(m0 + l16) * K3H_ + 2 * half;
    // B: column (n0+l16) of W1, starting at row offset 2*half
    const float* bcol = W1 + (size_t)(2 * half) * HID_ + (n0 + l16);

    v8f c = {};
    for (int k = 0; k < K3H_; k += 4) {
        v2f a, bb;
        a.x  = arow[k];
        a.y  = arow[k + 1];
        bb.x = bcol[(size_t)k * HID_];
        bb.y = bcol[(size_t)k * HID_ + HID_];
        // 8 args: (neg_a, A, neg_b, B, c_mod, C, reuse_a, reuse_b)
        c = __builtin_amdgcn_wmma_f32_16x16x4_f32(
                false, a, false, bb, (short)0, c, false, false);
    }

    // Fused epilogue: +b1, inference batchnorm, LeakyReLU(0.01).
    const int n = n0 + l16;
    const float scale = gamma[n] * rsqrtf(rvar[n] + EPS_);
    const float shift = beta[n] - rmean[n] * scale;
    const float bias1 = b1[n];
#pragma unroll
    for (int i = 0; i < 8; ++i) {
        const int m = m0 + i + 8 * half;
        float v = c[i] + bias1;
        v = v * scale + shift;
        v = (v >= 0.0f) ? v : 0.01f * v;
        h[(size_t)m * HID_ + n] = v;
    }
}

// ---------------------------------------------------------------------------
// Phase 3: out = h @ W2 + b2   (64x512 @ 512x3). 192 scalar dot products.
// ---------------------------------------------------------------------------
__global__ void coref_head2(const float* __restrict__ h,
                            const float* __restrict__ W2,
                            const float* __restrict__ b2,
                            float* __restrict__ out) {
    const int t = blockIdx.x * blockDim.x + threadIdx.x;
    if (t >= B_ * 3) return;
    const int row = t / 3;
    const int col = t % 3;
    const float* hr = h + (size_t)row * HID_;
    float acc = b2[col];
    for (int k = 0; k < HID_; ++k)
        acc = fmaf(hr[k], W2[k * 3 + col], acc);
    out[t] = acc;
}

extern "C" void kernel_launch(void* const* d_in, const int* in_sizes, int n_in,
                              void* d_out, int out_size, void* d_ws, size_t ws_size,
                              hipStream_t stream) {
    const float* bert    = (const float*)d_in[0];
    const int*   offsets = (const int*)d_in[1];
    const float* W1      = (const float*)d_in[2];
    const float* b1      = (const float*)d_in[3];
    const float* gamma   = (const float*)d_in[4];
    const float* beta    = (const float*)d_in[5];
    const float* rmean   = (const float*)d_in[6];
    const float* rvar    = (const float*)d_in[7];
    const float* W2      = (const float*)d_in[8];
    const float* b2      = (const float*)d_in[9];
    float*       out     = (float*)d_out;

    // Workspace layout: x (B x 3H fp32) then h (B x HID fp32). < 1 MB total.
    float* x = (float*)d_ws;
    float* h = x + (size_t)B_ * K3H_;

    // Phase 1: one block per batch, 256 threads (8 waves), float4 per thread.
    coref_span_gather<<<B_, 256, 0, stream>>>(bert, offsets, x);

    // Phase 2: 128 wave32 tiles -> 32 blocks x 128 threads (4 waves each).
    coref_gemm1_wmma<<<32, 128, 0, stream>>>(x, W1, b1, gamma, beta,
                                             rmean, rvar, h);

    // Phase 3: single small block.
    coref_head2<<<1, 192, 0, stream>>>(h, W2, b2, out);
}